// PhongRender_18726057411149
// MI455X (gfx1250) — compile-verified
//
#include <hip/hip_runtime.h>
#include <math.h>

typedef __attribute__((ext_vector_type(2))) float v2f;
typedef __attribute__((ext_vector_type(8))) float v8f;

// Problem constants from the reference harness (B=1).
#define RH 160
#define RW 160

// ---------------------------------------------------------------------------
// Kernel 1: per-face preprocessing.
//  - camera transform + perspective projection of the 3 vertices
//  - face normal (-> normal1 output, normalized)
//  - affine edge-function coefficients:  w = A*px + B*py + C
//    (invalid faces: |denom|<=1e-8 or nz<=0 get w0 == -1 everywhere -> never inside)
//  - z-plane coefficients: z = Az*px + Bz*py + Cz
//  - per-face shading data: p0,p1,p2 (eye vec), normal, uv0,uv1,uv2
// ---------------------------------------------------------------------------
__global__ void setup_faces_kernel(
    const float* __restrict__ points, const int* __restrict__ faces,
    const float* __restrict__ rot, const float* __restrict__ cpos,
    const float* __restrict__ proj, const float* __restrict__ uv,
    int F,
    float* __restrict__ coef,        // F x 12: [A0,B0,C0,0, A1,B1,C1,0, Az,Bz,Cz,0]
    float* __restrict__ fdata,       // F x 18: p0(3) p1(3) p2(3) n(3) uv0(2) uv1(2) uv2(2)
    float* __restrict__ normal1_out) // F x 3
{
    int f = blockIdx.x * blockDim.x + threadIdx.x;
    if (f >= F) return;

    float r00=rot[0],r01=rot[1],r02=rot[2];
    float r10=rot[3],r11=rot[4],r12=rot[5];
    float r20=rot[6],r21=rot[7],r22=rot[8];
    float cx=cpos[0],cy=cpos[1],cz=cpos[2];
    float pj0=proj[0],pj1=proj[1],pj2=proj[2];

    float P3[3][3], X[3], Y[3], UV[3][2];
    #pragma unroll
    for (int v = 0; v < 3; ++v) {
        int idx = faces[f*3 + v];
        float q0 = points[idx*3+0] - cx;
        float q1 = points[idx*3+1] - cy;
        float q2 = points[idx*3+2] - cz;
        float px_ = r00*q0 + r01*q1 + r02*q2;
        float py_ = r10*q0 + r11*q1 + r12*q2;
        float pz_ = r20*q0 + r21*q1 + r22*q2;
        P3[v][0]=px_; P3[v][1]=py_; P3[v][2]=pz_;
        float w = pz_*pj2;
        X[v] = (px_*pj0)/w;
        Y[v] = (py_*pj1)/w;
        UV[v][0] = uv[idx*2+0];
        UV[v][1] = uv[idx*2+1];
    }

    // normal = cross(p1-p0, p2-p0)
    float e1x=P3[1][0]-P3[0][0], e1y=P3[1][1]-P3[0][1], e1z=P3[1][2]-P3[0][2];
    float e2x=P3[2][0]-P3[0][0], e2y=P3[2][1]-P3[0][1], e2z=P3[2][2]-P3[0][2];
    float nx = e1y*e2z - e1z*e2y;
    float ny = e1z*e2x - e1x*e2z;
    float nz = e1x*e2y - e1y*e2x;

    // normal1 output: x / (sqrt(sum+1e-8) + 1e-15)
    float nl = sqrtf(nx*nx + ny*ny + nz*nz + 1e-8f) + 1e-15f;
    normal1_out[f*3+0] = nx/nl;
    normal1_out[f*3+1] = ny/nl;
    normal1_out[f*3+2] = nz/nl;

    float x0=X[0],y0=Y[0],x1=X[1],y1=Y[1],x2=X[2],y2=Y[2];
    float denom = (y1-y2)*(x0-x2) + (x2-x1)*(y0-y2);

    float A0,B0,C0,A1,B1,C1,Az,Bz,Cz;
    if (fabsf(denom) > 1e-8f && nz > 0.0f) {
        float inv = 1.0f/denom;
        A0 = (y1-y2)*inv;  B0 = (x2-x1)*inv;  C0 = -(A0*x2 + B0*y2);
        A1 = (y2-y0)*inv;  B1 = (x0-x2)*inv;  C1 = -(A1*x2 + B1*y2);
        float z0=P3[0][2], z1=P3[1][2], z2v=P3[2][2];
        float d0 = z0 - z2v, d1 = z1 - z2v;
        Az = A0*d0 + A1*d1;
        Bz = B0*d0 + B1*d1;
        Cz = C0*d0 + C1*d1 + z2v;
    } else {
        A0=0.f;B0=0.f;C0=-1.f;  A1=0.f;B1=0.f;C1=-1.f;  Az=0.f;Bz=0.f;Cz=0.f;
    }

    float* cf = coef + f*12;
    cf[0]=A0; cf[1]=B0; cf[2]=C0;  cf[3]=0.f;
    cf[4]=A1; cf[5]=B1; cf[6]=C1;  cf[7]=0.f;
    cf[8]=Az; cf[9]=Bz; cf[10]=Cz; cf[11]=0.f;

    float* fd = fdata + f*18;
    fd[0]=P3[0][0]; fd[1]=P3[0][1]; fd[2]=P3[0][2];
    fd[3]=P3[1][0]; fd[4]=P3[1][1]; fd[5]=P3[1][2];
    fd[6]=P3[2][0]; fd[7]=P3[2][1]; fd[8]=P3[2][2];
    fd[9]=nx; fd[10]=ny; fd[11]=nz;
    fd[12]=UV[0][0]; fd[13]=UV[0][1];
    fd[14]=UV[1][0]; fd[15]=UV[1][1];
    fd[16]=UV[2][0]; fd[17]=UV[2][1];
}

// ---------------------------------------------------------------------------
// Kernel 2: WMMA rasterizer. One wave32 handles 32 pixels (two 16-pixel B
// fragments) vs all F faces. Per 16-face tile: 3 coefficient loads feed
// 6x V_WMMA_F32_16X16X4_F32 (w0,w1,z for each pixel group), then min3-based
// inside-test + argmax. A-loads are software double-buffered so the
// s_wait_loadcnt for tile t+1 overlaps a full tile of WMMA+VALU work.
//
// Fragment layouts (ISA 7.12.2, f32):
//   A (16x4): lane<16 -> M=lane,    K=0,1 in V0,V1  (we store A,B there)
//             lane>=16-> M=lane-16, K=2,3 in V0,V1  (we store C,0 there)
//   B (4x16): lane<16 -> N=lane,    K=0,1 = [px,py]
//             lane>=16-> N=lane-16, K=2,3 = [1, 0]
//   D (16x16): lane l holds pixel N=l&15, faces M=(l>=16?8:0)+i in comp i.
// ---------------------------------------------------------------------------
__device__ __forceinline__ void tile_argmax(
    const v8f& w0, const v8f& w1, const v8f& zz, int fbase,
    float& bestz, int& bestf, float NEGINF)
{
    #pragma unroll
    for (int i = 0; i < 8; ++i) {
        float W0 = w0[i], W1 = w1[i];
        float W2 = 1.0f - W0 - W1;
        float m  = fminf(fminf(W0, W1), W2);   // -> v_min3_f32
        float zv = (m >= 0.0f) ? zz[i] : NEGINF;
        if (zv > bestz) { bestz = zv; bestf = fbase + i; }  // strict >: keep first max
    }
}

__global__ void raster_kernel(
    const float* __restrict__ coef, int F, int W, int H,
    int* __restrict__ bestface)
{
    const int HW = W * H;
    const int lane = threadIdx.x & 31;
    const int wid  = blockIdx.x * (blockDim.x >> 5) + (threadIdx.x >> 5);
    const int pixBase = wid * 32;          // 32 pixels per wave
    if (pixBase >= HW) return;             // wave-uniform: EXEC stays all-ones

    const int n  = lane & 15;              // pixel-in-group / face-in-tile
    const int hi = lane >> 4;              // K-half selector

    // pixel group A: pixBase + n ; pixel group B: pixBase + 16 + n
    const int pixA = pixBase + n;
    const int pixB = pixBase + 16 + n;
    int rowA = pixA / W, colA = pixA - rowA * W;
    int rowB = pixB / W, colB = pixB - rowB * W;
    const float sx = 2.0f / (float)W, sy = 2.0f / (float)H;
    float pxA = ((float)colA + 0.5f) * sx - 1.0f;
    float pyA = 1.0f - ((float)rowA + 0.5f) * sy;
    float pxB = ((float)colB + 0.5f) * sx - 1.0f;
    float pyB = 1.0f - ((float)rowB + 0.5f) * sy;

    v2f bA, bB;
    bA.x = hi ? 1.0f : pxA;  bA.y = hi ? 0.0f : pyA;
    bB.x = hi ? 1.0f : pxB;  bB.y = hi ? 0.0f : pyB;

    const float NEGINF = -__builtin_inff();
    float bzA = NEGINF, bzB = NEGINF;
    int   bfA = 0,      bfB = 0;

    const int ntiles = F >> 4;

    // prologue: load tile 0's A fragments
    const float* cp0 = coef + (size_t)n * 12 + (hi << 1);
    v2f a0 = *(const v2f*)(cp0 + 0);
    v2f a1 = *(const v2f*)(cp0 + 4);
    v2f az = *(const v2f*)(cp0 + 8);

    for (int t = 0; t < ntiles; ++t) {
        // issue next tile's loads before consuming current fragments
        const int tn = (t + 1 < ntiles) ? (t + 1) : t;
        const float* cpn = coef + (size_t)((tn << 4) + n) * 12 + (hi << 1);
        v2f na0 = *(const v2f*)(cpn + 0);
        v2f na1 = *(const v2f*)(cpn + 4);
        v2f naz = *(const v2f*)(cpn + 8);

        v8f zero = {};
        v8f w0A = __builtin_amdgcn_wmma_f32_16x16x4_f32(false, a0, false, bA, (short)0, zero, false, false);
        v8f w1A = __builtin_amdgcn_wmma_f32_16x16x4_f32(false, a1, false, bA, (short)0, zero, false, false);
        v8f zzA = __builtin_amdgcn_wmma_f32_16x16x4_f32(false, az, false, bA, (short)0, zero, false, false);
        v8f w0B = __builtin_amdgcn_wmma_f32_16x16x4_f32(false, a0, false, bB, (short)0, zero, false, false);
        v8f w1B = __builtin_amdgcn_wmma_f32_16x16x4_f32(false, a1, false, bB, (short)0, zero, false, false);
        v8f zzB = __builtin_amdgcn_wmma_f32_16x16x4_f32(false, az, false, bB, (short)0, zero, false, false);

        const int fbase = (t << 4) + (hi << 3);
        tile_argmax(w0A, w1A, zzA, fbase, bzA, bfA, NEGINF);
        tile_argmax(w0B, w1B, zzB, fbase, bzB, bfB, NEGINF);

        a0 = na0; a1 = na1; az = naz;
    }

    // combine the two half-wave face sets (wave32)
    {
        float oz = __shfl_xor(bzA, 16, 32);
        int   of = __shfl_xor(bfA, 16, 32);
        if (oz > bzA || (oz == bzA && of < bfA)) { bzA = oz; bfA = of; }
    }
    {
        float oz = __shfl_xor(bzB, 16, 32);
        int   of = __shfl_xor(bfB, 16, 32);
        if (oz > bzB || (oz == bzB && of < bfB)) { bzB = oz; bfB = of; }
    }

    if (lane < 16) {
        bestface[pixA] = (bzA > NEGINF) ? bfA : -1;
        if (pixB < HW) bestface[pixB] = (bzB > NEGINF) ? bfB : -1;
    }
}

// ---------------------------------------------------------------------------
// Kernel 3: per-pixel interpolation + Phong shading + nearest texel fetch.
// ---------------------------------------------------------------------------
__global__ void shade_kernel(
    const int* __restrict__ bestface,
    const float* __restrict__ coef, const float* __restrict__ fdata,
    const float* __restrict__ texture, const float* __restrict__ light,
    const float* __restrict__ material, const float* __restrict__ shininess,
    int W, int H, int T,
    float* __restrict__ imrender, float* __restrict__ improb)
{
    int p = blockIdx.x * blockDim.x + threadIdx.x;
    const int HW = W * H;
    if (p >= HW) return;

    int bf = bestface[p];
    if (bf < 0) {
        imrender[p*3+0] = 0.f; imrender[p*3+1] = 0.f; imrender[p*3+2] = 0.f;
        improb[p] = 0.f;
        return;
    }
    improb[p] = 1.0f;

    int row = p / W, col = p - row * W;
    float px = ((float)col + 0.5f) * (2.0f / (float)W) - 1.0f;
    float py = 1.0f - ((float)row + 0.5f) * (2.0f / (float)H);

    const float* c = coef + bf * 12;
    float w0 = c[0]*px + c[1]*py + c[2];
    float w1 = c[4]*px + c[5]*py + c[6];
    float w2 = 1.0f - w0 - w1;

    const float* fd = fdata + bf * 18;
    // eye = -(w0*p0 + w1*p1 + w2*p2), normalized
    float ex = -(w0*fd[0] + w1*fd[3] + w2*fd[6]);
    float ey = -(w0*fd[1] + w1*fd[4] + w2*fd[7]);
    float ez = -(w0*fd[2] + w1*fd[5] + w2*fd[8]);
    float el = sqrtf(ex*ex + ey*ey + ez*ez + 1e-8f) + 1e-15f;
    ex /= el; ey /= el; ez /= el;

    // interpolated normal = face normal (weights sum to 1), normalized
    float nx = fd[9], ny = fd[10], nz = fd[11];
    float nl = sqrtf(nx*nx + ny*ny + nz*nz + 1e-8f) + 1e-15f;
    nx /= nl; ny /= nl; nz /= nl;

    float uvx = w0*fd[12] + w1*fd[14] + w2*fd[16];
    float uvy = w0*fd[13] + w1*fd[15] + w2*fd[17];

    // normalized light direction
    float lx = light[0], ly = light[1], lz = light[2];
    float ll = sqrtf(lx*lx + ly*ly + lz*lz + 1e-8f) + 1e-15f;
    lx /= ll; ly /= ll; lz /= ll;

    float cosT = nx*lx + ny*ly + nz*lz;
    cosT = fminf(fmaxf(cosT, 0.0f), 1.0f);
    float rx = -lx + 2.0f*cosT*nx;
    float ry = -ly + 2.0f*cosT*ny;
    float rz = -lz + 2.0f*cosT*nz;
    float cosA = rx*ex + ry*ey + rz*ez;
    cosA = fminf(fmaxf(cosA, 1e-5f), 1.0f);
    cosA = powf(cosA, shininess[0]);

    // nearest texel: ix = clamp(rint(frac(u)*T - 0.5)), iy = clamp(rint(T*(1-frac(v)) - 0.5))
    float fu = uvx - floorf(uvx);
    float fv = uvy - floorf(uvy);
    float Tf = (float)T;
    float ixf = fminf(fmaxf(rintf(fu*Tf - 0.5f), 0.0f), Tf - 1.0f);
    float iyf = fminf(fmaxf(rintf(Tf - fv*Tf - 0.5f), 0.0f), Tf - 1.0f);
    int ix = (int)ixf, iy = (int)iyf;
    int tt = T * T;
    float t0 = texture[0*tt + iy*T + ix];
    float t1 = texture[1*tt + iy*T + ix];
    float t2 = texture[2*tt + iy*T + ix];

    float c0 = (material[0] + material[3]*cosT) * t0 + material[6]*cosA;
    float c1 = (material[1] + material[4]*cosT) * t1 + material[7]*cosA;
    float c2 = (material[2] + material[5]*cosT) * t2 + material[8]*cosA;
    imrender[p*3+0] = fminf(fmaxf(c0, 0.0f), 1.0f);
    imrender[p*3+1] = fminf(fmaxf(c1, 0.0f), 1.0f);
    imrender[p*3+2] = fminf(fmaxf(c2, 0.0f), 1.0f);
}

// ---------------------------------------------------------------------------
extern "C" void kernel_launch(void* const* d_in, const int* in_sizes, int n_in,
                              void* d_out, int out_size, void* d_ws, size_t ws_size,
                              hipStream_t stream) {
    const float* points   = (const float*)d_in[0];
    const int*   faces    = (const int*)  d_in[1];
    const float* rot      = (const float*)d_in[2];
    const float* cpos     = (const float*)d_in[3];
    const float* proj     = (const float*)d_in[4];
    const float* uv       = (const float*)d_in[5];
    const float* texture  = (const float*)d_in[6];
    const float* light    = (const float*)d_in[7];
    const float* material = (const float*)d_in[8];
    const float* shin     = (const float*)d_in[9];

    const int F = in_sizes[1] / 3;
    int T = 1;
    { long tpix = (long)in_sizes[6] / 3;              // T*T
      while ((long)T * (long)T < tpix) ++T; }
    const int H = RH, W = RW;
    const int HW = H * W;

    // workspace layout
    float* coef  = (float*)d_ws;          // F*12
    float* fdata = coef + (size_t)F * 12; // F*18
    int*   bestf = (int*)(fdata + (size_t)F * 18);

    // output layout: imrender (HW*3) | improb (HW) | normal1 (F*3)
    float* out      = (float*)d_out;
    float* imrender = out;
    float* improb   = out + (size_t)HW * 3;
    float* normal1  = out + (size_t)HW * 4;

    setup_faces_kernel<<<(F + 255) / 256, 256, 0, stream>>>(
        points, faces, rot, cpos, proj, uv, F, coef, fdata, normal1);

    const int nwaves  = (HW + 31) / 32;   // 32 pixels per wave32
    const int nblocks = (nwaves + 3) / 4; // 128 threads = 4 waves per block
    raster_kernel<<<nblocks, 128, 0, stream>>>(coef, F, W, H, bestf);

    shade_kernel<<<(HW + 255) / 256, 256, 0, stream>>>(
        bestf, coef, fdata, texture, light, material, shin, W, H, T,
        imrender, improb);
}